// LogMarginalLikelihood_21131239096654
// MI455X (gfx1250) — compile-verified
//
#include <hip/hip_runtime.h>
#include <math.h>

// ---------------------------------------------------------------------------
// GP log-marginal-likelihood via blocked Cholesky on gfx1250 (MI455X).
//   A = theta_f*exp(-d2/(2 l^2)) + (jitter + theta_n^2) I      (N x N, fp32)
//   A = L L^T  (blocked right-looking Cholesky, SYRK via V_WMMA_F32_16X16X4_F32)
//   lml = 0.5*sum_d ||L^{-1} y_d||^2 + sum_i log L_ii + 0.5*n*log(2*pi)
//
// MI455X reasoning: A is 64 MB -> lives in the 192 MB L2 after the build pass,
// so HBM (23.3 TB/s) is irrelevant for the factorization; the SYRK trailing
// update (~2/3 of N^3/3 FLOPs) must be math/L2 balanced. Precision (jitter
// 3e-7 in a logdet/inverse) mandates the full-fp32 matrix op
// V_WMMA_F32_16X16X4_F32 rather than bf16/fp16/fp8 WMMA. To lift arithmetic
// intensity off L2 (~4 -> ~16 FLOP/B) each workgroup (4 waves) computes a
// 64x64 output tile with both 64x64 panel slabs staged in LDS (34 KB of the
// 320 KB WGP pool); each wave owns a 16x64 strip and reuses its A-fragment
// across four B-fragments (4 WMMA per K-step, fragments via ds_load_b64).
// ---------------------------------------------------------------------------

#define NBLK 64           // Cholesky panel width (divides 4096)
#define TILE 64           // SYRK output tile per workgroup
#define LDSP (NBLK + 4)   // LDS row stride: keeps float4 stores 16B-aligned

typedef __attribute__((ext_vector_type(2))) float v2f;
typedef __attribute__((ext_vector_type(8))) float v8f;

// ---- 1) Build A ------------------------------------------------------------
__global__ void build_A_kernel(float* __restrict__ A, const float* __restrict__ t,
                               const float* __restrict__ thf, const float* __restrict__ thl,
                               const float* __restrict__ thn, int N) {
  long long idx = (long long)blockIdx.x * blockDim.x + threadIdx.x;
  long long total = (long long)N * N;
  if (idx >= total) return;
  int i = (int)(idx / N);
  int j = (int)(idx - (long long)i * N);
  float d = t[i] - t[j];
  float l = thl[0];
  float v = thf[0] * expf(-(d * d) / (2.0f * l * l));
  if (i == j) v += 3e-7f + thn[0] * thn[0];
  A[idx] = v;
}

// ---- 2a) Factor the NBLK x NBLK diagonal block in LDS (one workgroup) ------
__global__ void __launch_bounds__(NBLK) chol_diag_kernel(float* __restrict__ A, int N, int kb) {
  __shared__ float Lb[NBLK][NBLK + 1];
  int j = threadIdx.x;                   // this thread owns row j of the block
  for (int i = 0; i < NBLK; ++i) Lb[j][i] = A[(size_t)(kb + j) * N + kb + i];
  __syncthreads();
  for (int k = 0; k < NBLK; ++k) {
    if (j == k) Lb[k][k] = sqrtf(Lb[k][k]);
    __syncthreads();
    if (j > k) Lb[j][k] /= Lb[k][k];
    __syncthreads();
    if (j > k) {
      float ljk = Lb[j][k];
      for (int i = k + 1; i <= j; ++i) Lb[j][i] -= ljk * Lb[i][k];  // lower only
    }
    __syncthreads();
  }
  for (int i = 0; i < NBLK; ++i) A[(size_t)(kb + j) * N + kb + i] = Lb[j][i];
}

// ---- 2b) TRSM: L21 = A21 * L11^{-T}, one row per thread, L11 in LDS --------
__global__ void __launch_bounds__(128) chol_trsm_kernel(float* __restrict__ A, int N, int kb) {
  __shared__ float Lb[NBLK][NBLK + 1];
  int t = threadIdx.x;
  for (int idx = t; idx < NBLK * NBLK; idx += 128) {
    int r = idx / NBLK, c = idx - r * NBLK;
    Lb[r][c] = A[(size_t)(kb + r) * N + kb + c];
  }
  __syncthreads();
  int row = kb + NBLK + blockIdx.x * 128 + t;
  if (row >= N) return;                    // after last barrier: safe
  float x[NBLK];
  const size_t base = (size_t)row * N + kb;
  for (int jj = 0; jj < NBLK; ++jj) {      // forward substitution vs L11^T
    float s = A[base + jj];
    for (int m = 0; m < jj; ++m) s -= x[m] * Lb[jj][m];
    x[jj] = s / Lb[jj][jj];
  }
  for (int jj = 0; jj < NBLK; ++jj) A[base + jj] = x[jj];
}

// ---- 2c) SYRK trailing update, LDS-staged fp32 WMMA: A22 -= L21 * L21^T ----
// 128 threads = 4 waves per 64x64 output tile. Both panel slabs staged in LDS
// (float4 cooperative loads). Wave w computes rows [w*16, w*16+16) x 64 cols:
// one A-fragment per K-step reused across 4 B-fragments (ds_load_b64 pairs).
// Fragment layouts per ISA 7.12.2 (f32, 16x16x4):
//   A 16x4 : lanes 0-15 = M rows with K{0,1}; lanes 16-31 = K{2,3}
//   B 4x16 : lanes 0-15 = N cols with K{0,1}; lanes 16-31 = K{2,3}
//   C/D    : VGPR r -> M=r (lanes 0-15) / M=r+8 (lanes 16-31), N=lane&15
__global__ void __launch_bounds__(128) syrk_wmma_kernel(float* __restrict__ A, int N,
                                                        int kb, int off) {
  __shared__ float Ablk[TILE][LDSP];
  __shared__ float Bblk[TILE][LDSP];

  const int i0   = off + blockIdx.x * TILE;   // output tile row origin
  const int j0   = off + blockIdx.y * TILE;   // output tile col origin
  const int tid  = threadIdx.x;               // 0..127
  const int wave = tid >> 5;                  // 0..3
  const int lane = tid & 31;
  const int half = lane >> 4;                 // K-pair selector (0|1)
  const int l    = lane & 15;                 // row/col within 16-tile

  // Cooperative stage of both 64x64 slabs of the L21 panel (cols kb..kb+63).
  for (int p = tid; p < TILE * (NBLK / 4); p += 128) {
    const int r = p >> 4;                     // slab row
    const int c = (p & 15) << 2;              // slab col (float4 granule)
    const float4 va = *(const float4*)&A[(size_t)(i0 + r) * N + kb + c];
    const float4 vb = *(const float4*)&A[(size_t)(j0 + r) * N + kb + c];
    *(float4*)&Ablk[r][c] = va;
    *(float4*)&Bblk[r][c] = vb;
  }
  __syncthreads();

  const int ri = wave * 16;                   // this wave's row strip
  v8f acc0 = {}, acc1 = {}, acc2 = {}, acc3 = {};
#pragma unroll
  for (int k = 0; k < NBLK; k += 4) {
    const int ka = k + half * 2;
    v2f a, b0, b1, b2, b3;
    a.x  = Ablk[ri + l][ka];  a.y  = Ablk[ri + l][ka + 1];   // ds_load_b64
    b0.x = Bblk[ 0 + l][ka];  b0.y = Bblk[ 0 + l][ka + 1];
    b1.x = Bblk[16 + l][ka];  b1.y = Bblk[16 + l][ka + 1];
    b2.x = Bblk[32 + l][ka];  b2.y = Bblk[32 + l][ka + 1];
    b3.x = Bblk[48 + l][ka];  b3.y = Bblk[48 + l][ka + 1];
    // D = A x B + C ; 8 args: (neg_a, A, neg_b, B, c_mod, C, reuse_a, reuse_b)
    acc0 = __builtin_amdgcn_wmma_f32_16x16x4_f32(false, a, false, b0, (short)0, acc0, false, false);
    acc1 = __builtin_amdgcn_wmma_f32_16x16x4_f32(false, a, false, b1, (short)0, acc1, false, false);
    acc2 = __builtin_amdgcn_wmma_f32_16x16x4_f32(false, a, false, b2, (short)0, acc2, false, false);
    acc3 = __builtin_amdgcn_wmma_f32_16x16x4_f32(false, a, false, b3, (short)0, acc3, false, false);
  }

  // Writeback: subtract (f32 WMMA has no A/B negate). Each wave owns its strip.
  const int m0 = i0 + ri + half * 8;
#pragma unroll
  for (int r = 0; r < 8; ++r) {
    const size_t rowb = (size_t)(m0 + r) * N + j0 + l;
    A[rowb +  0] -= acc0[r];
    A[rowb + 16] -= acc1[r];
    A[rowb + 32] -= acc2[r];
    A[rowb + 48] -= acc3[r];
  }
}

// ---- 3) Forward solves + logdet + final combine (single workgroup) ---------
__global__ void __launch_bounds__(256) solve_finalize_kernel(const float* __restrict__ A,
                                                             const float* __restrict__ traj,
                                                             const int* __restrict__ nptr,
                                                             float* __restrict__ out,
                                                             int N, int D) {
  __shared__ float z[4096];
  __shared__ float red[256];
  const int t = threadIdx.x;
  float trace = 0.0f;                        // only thread 0's copy is used

  for (int d = 0; d < D; ++d) {
    const float* y = traj + (size_t)d * N;
    for (int i = 0; i < N; ++i) {
      const float* Li = A + (size_t)i * N;
      float p = 0.0f;
      for (int m = t; m < i; m += 256) p += Li[m] * z[m];
      red[t] = p; __syncthreads();
      for (int s = 128; s > 0; s >>= 1) { if (t < s) red[t] += red[t + s]; __syncthreads(); }
      if (t == 0) z[i] = (y[i] - red[0]) / Li[i];
      __syncthreads();
    }
    float p = 0.0f;
    for (int m = t; m < N; m += 256) p += z[m] * z[m];
    red[t] = p; __syncthreads();
    for (int s = 128; s > 0; s >>= 1) { if (t < s) red[t] += red[t + s]; __syncthreads(); }
    if (t == 0) trace += red[0];
    __syncthreads();
  }

  float p = 0.0f;
  for (int m = t; m < N; m += 256) p += logf(A[(size_t)m * N + m]);
  red[t] = p; __syncthreads();
  for (int s = 128; s > 0; s >>= 1) { if (t < s) red[t] += red[t + s]; __syncthreads(); }
  if (t == 0) {
    float logdet = 2.0f * red[0];
    float lml = 0.5f * trace + 0.5f * logdet
              + 0.5f * (float)nptr[0] * logf(2.0f * 3.14159265358979323846f);
    out[0] = lml;
  }
}

// ---------------------------------------------------------------------------
extern "C" void kernel_launch(void* const* d_in, const int* in_sizes, int n_in,
                              void* d_out, int out_size, void* d_ws, size_t ws_size,
                              hipStream_t stream) {
  (void)n_in; (void)out_size; (void)ws_size;
  const float* traj = (const float*)d_in[0];   // [D, N] row-major
  const float* t    = (const float*)d_in[1];   // [N]
  const float* thf  = (const float*)d_in[2];
  const float* thl  = (const float*)d_in[3];
  const float* thn  = (const float*)d_in[4];
  const int*   npar = (const int*)d_in[5];

  const int N = in_sizes[1];                   // 4096
  const int D = in_sizes[0] / N;               // 4
  float* A = (float*)d_ws;                     // N*N*4 = 64 MB scratch

  // 1) Build kernel matrix
  {
    long long total = (long long)N * N;
    int threads = 256;
    unsigned blocks = (unsigned)((total + threads - 1) / threads);
    build_A_kernel<<<blocks, threads, 0, stream>>>(A, t, thf, thl, thn, N);
  }

  // 2) Blocked right-looking Cholesky
  for (int kb = 0; kb < N; kb += NBLK) {
    chol_diag_kernel<<<1, NBLK, 0, stream>>>(A, N, kb);
    int rows = N - kb - NBLK;                  // multiple of NBLK (and of TILE)
    if (rows > 0) {
      chol_trsm_kernel<<<(rows + 127) / 128, 128, 0, stream>>>(A, N, kb);
      dim3 grid(rows / TILE, rows / TILE);
      syrk_wmma_kernel<<<grid, 128, 0, stream>>>(A, N, kb, kb + NBLK);
    }
  }

  // 3) Triangular solves + logdet + scalar combine
  solve_finalize_kernel<<<1, 256, 0, stream>>>(A, traj, npar, (float*)d_out, N, D);
}